// MessageLayer_90915867722551
// MI455X (gfx1250) — compile-verified
//
#include <hip/hip_runtime.h>

typedef __attribute__((ext_vector_type(2))) float v2f;
typedef __attribute__((ext_vector_type(8))) float v8f;

// ---------------------------------------------------------------------------
// Setup: build W[84][16] = sum_w weights[w] * CG_w laid out as the GEMM B
// matrix (feature index f = sh_idx*9 + src_idx, padded 81->84 rows; output
// column k padded 9->16). CG computed exactly per e3nn convention in double.
// ---------------------------------------------------------------------------

struct cpx { double re, im; };
static __device__ __forceinline__ cpx cmul(cpx a, cpx b) {
    cpx r; r.re = a.re * b.re - a.im * b.im; r.im = a.re * b.im + a.im * b.re; return r;
}

static __device__ __forceinline__ double factd(int n) {
    double r = 1.0;
    for (int i = 2; i <= n; ++i) r *= (double)i;
    return r;
}

static __device__ double su2_cg(int j1, int m1, int j2, int m2, int j3, int m3) {
    if (m3 != m1 + m2) return 0.0;
    int vmin = -j1 + j2 + m3; if (-j1 + m1 > vmin) vmin = -j1 + m1; if (vmin < 0) vmin = 0;
    int vmax = j2 + j3 + m1; if (j3 - j1 + j2 < vmax) vmax = j3 - j1 + j2; if (j3 + m3 < vmax) vmax = j3 + m3;
    double C = sqrt((2.0 * j3 + 1.0)
                    * factd(j3 + j1 - j2) * factd(j3 - j1 + j2) * factd(j1 + j2 - j3)
                    * factd(j3 + m3) * factd(j3 - m3)
                    / (factd(j1 + j2 + j3 + 1) * factd(j1 - m1) * factd(j1 + m1)
                       * factd(j2 - m2) * factd(j2 + m2)));
    double S = 0.0;
    for (int v = vmin; v <= vmax; ++v) {
        double sgn = ((v + j2 + m2) & 1) ? -1.0 : 1.0;
        S += sgn * factd(j2 + j3 + m1 - v) * factd(j1 - m1 + v)
             / (factd(v) * factd(j3 - j1 + j2 - v) * factd(j3 + m3 - v)
                * factd(v + j1 - j2 - m3));
    }
    return C * S;
}

static __device__ void qmat(int l, cpx q[5][5]) {
    for (int a = 0; a < 5; ++a)
        for (int b = 0; b < 5; ++b) { q[a][b].re = 0.0; q[a][b].im = 0.0; }
    double r2 = 1.0 / sqrt(2.0);
    for (int m = -l; m < 0; ++m) {
        q[l + m][l - m].re = r2;                 // q[l+m, l+|m|] = r2
        q[l + m][l + m].im = -r2;                // q[l+m, l-|m|] = -i*r2
    }
    q[l][l].re = 1.0;
    for (int m = 1; m <= l; ++m) {
        double s = (m & 1) ? -1.0 : 1.0;
        q[l + m][l + m].re = s * r2;             // (-1)^m * r2
        q[l + m][l - m].im = s * r2;             // i*(-1)^m * r2
    }
    cpx ph;
    switch (l & 3) {                              // (-i)^l
        case 0:  ph.re = 1.0;  ph.im = 0.0;  break;
        case 1:  ph.re = 0.0;  ph.im = -1.0; break;
        case 2:  ph.re = -1.0; ph.im = 0.0;  break;
        default: ph.re = 0.0;  ph.im = 1.0;  break;
    }
    for (int a = 0; a < 5; ++a)
        for (int b = 0; b < 5; ++b) q[a][b] = cmul(ph, q[a][b]);
}

static __device__ void so3_cg_dev(int l1, int l2, int l3, double out[5][5][5]) {
    int d1 = 2 * l1 + 1, d2 = 2 * l2 + 1, d3 = 2 * l3 + 1;
    double Cw[5][5][5];
    for (int i = 0; i < 5; ++i)
        for (int k = 0; k < 5; ++k)
            for (int n = 0; n < 5; ++n) Cw[i][k][n] = 0.0;
    for (int m1 = -l1; m1 <= l1; ++m1)
        for (int m2 = -l2; m2 <= l2; ++m2) {
            int m3 = m1 + m2;
            if (m3 >= -l3 && m3 <= l3)
                Cw[l1 + m1][l2 + m2][l3 + m3] = su2_cg(l1, m1, l2, m2, l3, m3);
        }
    cpx q1[5][5], q2[5][5], q3[5][5];
    qmat(l1, q1); qmat(l2, q2); qmat(l3, q3);
    // res[j][l][m] = sum_{i,k,n} q1[i][j] q2[k][l] conj(q3[n][m]) Cw[i][k][n]
    double nrm = 0.0;
    for (int j = 0; j < d1; ++j)
        for (int lI = 0; lI < d2; ++lI)
            for (int m = 0; m < d3; ++m) {
                double ar = 0.0;
                for (int i = 0; i < d1; ++i)
                    for (int k = 0; k < d2; ++k)
                        for (int n = 0; n < d3; ++n) {
                            double c = Cw[i][k][n];
                            if (c == 0.0) continue;
                            cpx t = cmul(q1[i][j], q2[k][lI]);
                            cpx c3; c3.re = q3[n][m].re; c3.im = -q3[n][m].im;
                            t = cmul(t, c3);
                            ar += t.re * c;      // real part only (imag ~ 0)
                        }
                out[j][lI][m] = ar;
                nrm += ar * ar;
            }
    nrm = sqrt(nrm);
    if (nrm > 0.0) {
        double inv = 1.0 / nrm;
        for (int j = 0; j < d1; ++j)
            for (int lI = 0; lI < d2; ++lI)
                for (int m = 0; m < d3; ++m) out[j][lI][m] *= inv;
    }
}

__global__ void build_W_kernel(const float* __restrict__ weights, float* __restrict__ W) {
    if (threadIdx.x != 0 || blockIdx.x != 0) return;
    for (int i = 0; i < 84 * 16; ++i) W[i] = 0.0f;
    const int OFFS[4] = {0, 1, 4, 9};
    // Combos in the exact python enumeration order
    const int CL[15] = {0, 0, 0, 1, 1, 1, 1, 1, 1, 2, 2, 2, 2, 2, 2};
    const int CN[15] = {0, 1, 2, 0, 1, 1, 1, 2, 2, 0, 1, 1, 2, 2, 2};
    const int CO[15] = {0, 1, 2, 1, 0, 1, 2, 1, 2, 2, 1, 2, 0, 1, 2};
    double C[5][5][5];
    for (int w = 0; w < 15; ++w) {
        int L = CL[w], ni = CN[w], oi = CO[w];
        so3_cg_dev(L, ni, oi, C);
        double wt = (double)weights[w];
        int dL = 2 * L + 1, dn = 2 * ni + 1, doo = 2 * oi + 1;
        for (int i = 0; i < dL; ++i)
            for (int j = 0; j < dn; ++j)
                for (int k = 0; k < doo; ++k) {
                    int f = (OFFS[L] + i) * 9 + (OFFS[ni] + j);   // 0..80
                    int col = OFFS[oi] + k;                        // 0..8
                    W[f * 16 + col] += (float)(wt * C[i][j][k]);
                }
    }
}

// ---------------------------------------------------------------------------
// Main edge kernel: per wave-tile of 16 edges, D[16x16] += A[16x4]*B[4x16]
// over 21 K-chunks via V_WMMA_F32_16X16X4_F32. A built in-register from
// sh(x)src outer product; B preloaded once; scatter-add via f32 atomics.
// ---------------------------------------------------------------------------

__global__ __launch_bounds__(256) void edge_kernel(
    const float* __restrict__ node, const float* __restrict__ Wmat,
    const float* __restrict__ sh0, const float* __restrict__ sh1,
    const float* __restrict__ sh2, const int* __restrict__ eidx,
    float* __restrict__ out, float* __restrict__ counts,
    int n_edges, int tiles_total)
{
    const int lane = threadIdx.x & 31;
    const int wv = threadIdx.x >> 5;
    const bool hihalf = lane >= 16;
    const int ncol = lane & 15;

    // Preload B operand registers: lane (half h, col n) needs rows 4c+2h, 4c+2h+1
    float bmat[42];
    {
        const int dk = hihalf ? 2 : 0;
#pragma unroll
        for (int c = 0; c < 21; ++c) {
            bmat[2 * c + 0] = Wmat[(4 * c + dk + 0) * 16 + ncol];
            bmat[2 * c + 1] = Wmat[(4 * c + dk + 1) * 16 + ncol];
        }
    }

    const int waves_per_block = blockDim.x >> 5;
    const int wave_stride = gridDim.x * waves_per_block;

    for (int tile = blockIdx.x * waves_per_block + wv; tile < tiles_total; tile += wave_stride) {
        const int e = tile * 16 + ncol;
        const bool valid = e < n_edges;
        const int ec = valid ? e : 0;

        const int src = eidx[ec];
        const int tgt = eidx[n_edges + ec];

        float s[9], x[9];
        s[0] = sh0[ec];
#pragma unroll
        for (int j = 0; j < 3; ++j) s[1 + j] = sh1[3 * ec + j];
#pragma unroll
        for (int j = 0; j < 5; ++j) s[4 + j] = sh2[5 * ec + j];
#pragma unroll
        for (int j = 0; j < 9; ++j) x[j] = node[9 * src + j];
        if (!valid) {
#pragma unroll
            for (int j = 0; j < 9; ++j) s[j] = 0.0f;
        }

        v8f acc = {};
#pragma unroll
        for (int c = 0; c < 21; ++c) {
            const int k0 = 4 * c;
            float l0 = (k0 + 0 < 81) ? s[(k0 + 0) / 9] * x[(k0 + 0) % 9] : 0.0f;
            float l1 = (k0 + 1 < 81) ? s[(k0 + 1) / 9] * x[(k0 + 1) % 9] : 0.0f;
            float h0 = (k0 + 2 < 81) ? s[(k0 + 2) / 9] * x[(k0 + 2) % 9] : 0.0f;
            float h1 = (k0 + 3 < 81) ? s[(k0 + 3) / 9] * x[(k0 + 3) % 9] : 0.0f;
            v2f A; A.x = hihalf ? h0 : l0; A.y = hihalf ? h1 : l1;
            v2f B; B.x = bmat[2 * c + 0]; B.y = bmat[2 * c + 1];
            acc = __builtin_amdgcn_wmma_f32_16x16x4_f32(
                false, A, false, B, (short)0, acc, false, false);
        }

        // Broadcast targets for the 8 rows this lane holds (all lanes execute)
        int tgts[8];
#pragma unroll
        for (int r = 0; r < 8; ++r) tgts[r] = __shfl(tgt, (hihalf ? 8 : 0) + r, 32);

        if (!hihalf && valid)
            __hip_atomic_fetch_add(&counts[tgt], 1.0f, __ATOMIC_RELAXED,
                                   __HIP_MEMORY_SCOPE_AGENT);

        if (ncol < 9) {
            const int mbase = hihalf ? 8 : 0;
#pragma unroll
            for (int r = 0; r < 8; ++r) {
                if (tile * 16 + mbase + r < n_edges)
                    __hip_atomic_fetch_add(&out[tgts[r] * 9 + ncol], acc[r],
                                           __ATOMIC_RELAXED, __HIP_MEMORY_SCOPE_AGENT);
            }
        }
    }
}

__global__ void init_kernel(float* __restrict__ out, float* __restrict__ counts,
                            int n_out, int n_nodes) {
    int stride = gridDim.x * blockDim.x;
    for (int k = blockIdx.x * blockDim.x + threadIdx.x; k < n_out; k += stride) out[k] = 0.0f;
    for (int k = blockIdx.x * blockDim.x + threadIdx.x; k < n_nodes; k += stride) counts[k] = 0.0f;
}

__global__ void norm_kernel(float* __restrict__ out, const float* __restrict__ counts,
                            int n_elems) {
    int i = blockIdx.x * blockDim.x + threadIdx.x;
    if (i < n_elems) {
        float c = counts[i / 9];
        out[i] /= fmaxf(c, 1.0f);
    }
}

extern "C" void kernel_launch(void* const* d_in, const int* in_sizes, int n_in,
                              void* d_out, int out_size, void* d_ws, size_t ws_size,
                              hipStream_t stream) {
    const float* node    = (const float*)d_in[0];
    const float* weights = (const float*)d_in[1];
    const float* sh0     = (const float*)d_in[2];
    const float* sh1     = (const float*)d_in[3];
    const float* sh2     = (const float*)d_in[4];
    const int*   eidx    = (const int*)d_in[5];
    float* out = (float*)d_out;

    const int n_edges = in_sizes[2];        // sh0 is [E,1]
    const int n_nodes = in_sizes[0] / 9;

    float* counts = (float*)d_ws;
    float* Wmat = counts + ((n_nodes + 255) & ~255);

    init_kernel<<<512, 256, 0, stream>>>(out, counts, out_size, n_nodes);
    build_W_kernel<<<1, 32, 0, stream>>>(weights, Wmat);

    const int tiles = (n_edges + 15) / 16;
    const int blocks = (tiles + 7) / 8;     // 8 waves / 256-thread block
    edge_kernel<<<blocks, 256, 0, stream>>>(node, Wmat, sh0, sh1, sh2, eidx,
                                            out, counts, n_edges, tiles);

    const int nelem = n_nodes * 9;
    norm_kernel<<<(nelem + 255) / 256, 256, 0, stream>>>(out, counts, nelem);
}